// QuantumAttention_781684048357
// MI455X (gfx1250) — compile-verified
//
#include <hip/hip_runtime.h>

typedef __attribute__((ext_vector_type(2))) float v2f;
typedef __attribute__((ext_vector_type(8))) float v8f;

constexpr int   B_DIM   = 4;
constexpr int   S_DIM   = 4096;
constexpr float EPS_C   = 1e-6f;
constexpr float DECAY   = 0.99f;   // HEBBIAN_DECAY
constexpr float HEBB    = 0.1f;    // HEBBIAN_STRENGTH
constexpr float CCLAMP  = 1.0f;    // CONNECTION_CLAMP
constexpr float MEMC    = 0.3f;    // MEMORY_COEF

// ---------------------------------------------------------------------------
// Pass 1: per-row adaptive threshold thr[b][i] = mean_i + 0.5*sqrt(var_i+EPS)
// One 256-thread block handles 256 rows of one batch; r_embed for the whole
// batch (4096 float2 = 32KB) is staged in LDS (WGP has 320KB available).
// sr[j] is wave-uniform in the inner loop -> broadcast read, conflict-free.
// ---------------------------------------------------------------------------
__global__ __launch_bounds__(256) void row_threshold_kernel(
    const float2* __restrict__ r, float* __restrict__ thr) {
  __shared__ float2 sr[S_DIM];
  const int chunksPerBatch = S_DIM / 256;               // 16
  const int b     = blockIdx.x / chunksPerBatch;
  const int chunk = blockIdx.x % chunksPerBatch;
  const float2* rb = r + (size_t)b * S_DIM;

  for (int j = threadIdx.x; j < S_DIM; j += 256) sr[j] = rb[j];
  __syncthreads();

  const int i = chunk * 256 + threadIdx.x;
  const float xi = sr[i].x, yi = sr[i].y;
  float sum = 0.f, sumsq = 0.f;
#pragma unroll 4
  for (int j = 0; j < S_DIM; ++j) {
    float d = fmaf(xi, sr[j].x, yi * sr[j].y);
    float s = fmaf(0.5f, d, 0.5f);                      // (dot + 1) * 0.5
    sum   += s;
    sumsq  = fmaf(s, s, sumsq);
  }
  const float inv = 1.0f / (float)S_DIM;
  float mean = sum * inv;
  float var  = fmaf(sumsq, inv, -mean * mean) + EPS_C;
  thr[(size_t)b * S_DIM + i] = fmaf(0.5f, sqrtf(var), mean);
}

// ---------------------------------------------------------------------------
// Pass 2: one wave32 per 16x16 output tile.
//  - Fully-above-diagonal tiles (~50% of traffic): no WMMA needed; re-map
//    lanes to contiguous float4 chunks -> b128 loads/stores (512B per wave
//    instruction) for pure decay+clamp streaming.
//  - Causal tiles: sim tile via one V_WMMA_F32_16X16X4_F32 (K=2 real data,
//    K=2..3 zero-padded), fused elementwise epilogue on the C layout
//    (VGPR k, lanes 0-15 -> M=k, N=lane; lanes 16-31 -> M=k+8, N=lane-16).
// ---------------------------------------------------------------------------
__global__ __launch_bounds__(256) void hebbian_tile_kernel(
    const float2* __restrict__ r, const float* __restrict__ conn,
    const float* __restrict__ thr, float* __restrict__ out,
    float* __restrict__ nconn) {
  const int wave   = threadIdx.x >> 5;
  const int lane   = threadIdx.x & 31;
  const int tile   = blockIdx.x * 8 + wave;             // tiles: B * 256 * 256
  const int b      = tile >> 16;
  const int ti     = (tile >> 8) & 255;
  const int tj     = tile & 255;
  const int row0   = ti << 4;
  const int col0   = tj << 4;

  if (col0 > row0 + 15) {
    // Entirely above the diagonal: score == 0, memory effect masked.
    // out = 0, new_conn = clamp(decay * conn). Lane mapping is free here:
    // use contiguous float4 chunks (16B/lane, 512B/wave instruction).
    const size_t tbase = ((size_t)b * S_DIM + row0) * S_DIM + col0;
    const float4 zero4 = make_float4(0.f, 0.f, 0.f, 0.f);
#pragma unroll
    for (int it = 0; it < 2; ++it) {
      const int e  = it * 32 + lane;                    // 0..63 float4 chunks
      const int m  = e >> 2;                            // row 0..15
      const int c4 = (e & 3) << 2;                      // col 0,4,8,12
      const size_t idx = tbase + (size_t)m * S_DIM + c4;
      float4 cv = *(const float4*)(conn + idx);
      float4 nc;
      nc.x = fminf(fmaxf(cv.x * DECAY, -CCLAMP), CCLAMP);
      nc.y = fminf(fmaxf(cv.y * DECAY, -CCLAMP), CCLAMP);
      nc.z = fminf(fmaxf(cv.z * DECAY, -CCLAMP), CCLAMP);
      nc.w = fminf(fmaxf(cv.w * DECAY, -CCLAMP), CCLAMP);
      *(float4*)(out + idx)   = zero4;
      *(float4*)(nconn + idx) = nc;
    }
    return;
  }

  const int hi     = lane >> 4;                         // 0 or 1
  const int lanelo = lane & 15;                         // N within tile
  const size_t cbase =
      ((size_t)b * S_DIM + row0) * S_DIM + col0 + lanelo;

  // A (16x4, rows of this tile) and B (4x16, cols of this tile):
  // lanes 0-15 hold K=0,1 (the real D=2 data); lanes 16-31 hold K=2,3 (zero).
  v2f a = {0.f, 0.f};
  v2f bm = {0.f, 0.f};
  if (lane < 16) {
    float2 ra = r[(size_t)b * S_DIM + row0 + lane];
    float2 rc = r[(size_t)b * S_DIM + col0 + lane];
    a.x = ra.x;  a.y = ra.y;
    bm.x = rc.x; bm.y = rc.y;
  }

  v8f c = {};
  // D = A x B + 0 : 16x16 dot-product tile in 8 VGPRs.
  c = __builtin_amdgcn_wmma_f32_16x16x4_f32(
      /*neg_a=*/false, a, /*neg_b=*/false, bm,
      /*c_mod=*/(short)0, c, /*reuse_a=*/false, /*reuse_b=*/false);

  // Row thresholds: lane l (<16) loads thr[row0+l]; broadcast per row via shfl.
  float thr_lane = (lane < 16) ? thr[(size_t)b * S_DIM + row0 + lane] : 0.f;

#pragma unroll
  for (int k = 0; k < 8; ++k) {
    const int m = k + 8 * hi;                           // row within tile
    const float tk = __shfl(thr_lane, m, 32);
    const size_t idx = cbase + (size_t)m * S_DIM;

    float sim = fmaf(c[k], 0.5f, 0.5f);                 // (dot + 1) * 0.5
    const bool causal = (col0 + lanelo) <= (row0 + m);
    const bool valid  = (sim >= tk) && causal;
    float score = valid ? fminf(fmaxf(sim, 0.f), 1.f) : 0.f;

    float nc = conn[idx] * DECAY;
    if (valid && (score > 0.1f)) nc = fmaf(HEBB, score, nc);
    nc = fminf(fmaxf(nc, -CCLAMP), CCLAMP);

    float o = score + (causal ? MEMC * nc : 0.f);
    o = fminf(fmaxf(o, 0.f), 1.f);

    out[idx]   = o;
    nconn[idx] = nc;
  }
}

// ---------------------------------------------------------------------------
// Launch: pass1 -> d_ws thresholds (B*S floats = 64KB), pass2 streams the
// [B,S,S] volume once (read conn, write out + new_conn = 768MB @ 23.3TB/s
// => ~33us floor; compute is ~2 GFLOP and rides along for free).
// ---------------------------------------------------------------------------
extern "C" void kernel_launch(void* const* d_in, const int* in_sizes, int n_in,
                              void* d_out, int out_size, void* d_ws, size_t ws_size,
                              hipStream_t stream) {
  (void)in_sizes; (void)n_in; (void)out_size; (void)ws_size;
  const float2* r    = (const float2*)d_in[0];  // r_embed [B,S,2] f32
  const float*  conn = (const float*)d_in[1];   // connection_strength [B,S,S]
  float* out   = (float*)d_out;                              // first output
  float* nconn = out + (size_t)B_DIM * S_DIM * S_DIM;        // second output
  float* thr   = (float*)d_ws;                               // B*S floats

  row_threshold_kernel<<<B_DIM * (S_DIM / 256), 256, 0, stream>>>(r, thr);

  const int tiles  = B_DIM * (S_DIM / 16) * (S_DIM / 16);    // 262144
  const int blocks = tiles / 8;                              // 8 waves/block
  hebbian_tile_kernel<<<blocks, 256, 0, stream>>>(r, conn, thr, out, nconn);
}